// Xtoy_19713899889388
// MI455X (gfx1250) — compile-verified
//
#include <hip/hip_runtime.h>

// ---------------------------------------------------------------------------
// Problem constants (from reference): X[64,16384,128] -> z[64,512] -> out[64,256]
// ---------------------------------------------------------------------------
#define BS      64
#define NN      16384
#define DX      128
#define DY      256
#define DZ      (4 * DX)            // 512
#define NSPLIT  64
#define ROWS    (NN / NSPLIT)       // 256 rows per partial block
#define PART    (BS * NSPLIT * DX)  // elements per partial array (524288)

typedef __attribute__((ext_vector_type(2))) float v2f;
typedef __attribute__((ext_vector_type(4))) float v4f;
typedef __attribute__((ext_vector_type(8))) float v8f;

#define FLT_BIG 3.402823466e+38f

// ---------------------------------------------------------------------------
// Kernel 1: partial stats over a 256-row slab. One wave per block; lane L owns
// columns 4L..4L+3, so each row is a single global_load_b128 per lane
// (32 x 16B = 512B contiguous). X is 512MB read once (>L2=192MB, no reuse)
// -> non-temporal loads. 4096 waves total, 8 outstanding b128 loads each:
// enough MLP to sit at the 23.3 TB/s HBM roofline, which bounds this problem.
// ---------------------------------------------------------------------------
__global__ void __launch_bounds__(32)
partial_reduce_kernel(const float* __restrict__ X, float* __restrict__ ws)
{
    const int b    = blockIdx.x;
    const int s    = blockIdx.y;
    const int lane = threadIdx.x;            // 0..31

    const v4f* __restrict__ p =
        (const v4f*)(X + ((size_t)b * NN + (size_t)s * ROWS) * DX) + lane;

    v4f sum = {0.0f, 0.0f, 0.0f, 0.0f};
    v4f sq  = {0.0f, 0.0f, 0.0f, 0.0f};
    v4f mn  = { FLT_BIG,  FLT_BIG,  FLT_BIG,  FLT_BIG};
    v4f mx  = {-FLT_BIG, -FLT_BIG, -FLT_BIG, -FLT_BIG};

#pragma unroll 8
    for (int i = 0; i < ROWS; ++i) {
        v4f x = __builtin_nontemporal_load(p + (size_t)i * (DX / 4));
        sum += x;
        sq  += x * x;
        mn.x = fminf(mn.x, x.x); mn.y = fminf(mn.y, x.y);
        mn.z = fminf(mn.z, x.z); mn.w = fminf(mn.w, x.w);
        mx.x = fmaxf(mx.x, x.x); mx.y = fmaxf(mx.y, x.y);
        mx.z = fmaxf(mx.z, x.z); mx.w = fmaxf(mx.w, x.w);
    }

    const size_t o = ((size_t)(b * NSPLIT + s)) * DX + 4 * lane;
    *(v4f*)(ws + o)            = sum;
    *(v4f*)(ws + PART + o)     = sq;
    *(v4f*)(ws + 2 * PART + o) = mn;
    *(v4f*)(ws + 3 * PART + o) = mx;
}

// ---------------------------------------------------------------------------
// Kernel 2: combine 64 partials per (b,d); emit z[b, 4*dx] =
// [mean | min | max | std(ddof=1)]. 8192 threads total.
// ---------------------------------------------------------------------------
__global__ void __launch_bounds__(128)
combine_stats_kernel(const float* __restrict__ ws, float* __restrict__ z)
{
    const int t = blockIdx.x * blockDim.x + threadIdx.x;  // 0 .. BS*DX-1
    const int b = t / DX;
    const int d = t - b * DX;

    float sum = 0.0f;
    float sq  = 0.0f;
    float mn  =  FLT_BIG;
    float mx  = -FLT_BIG;

#pragma unroll 8
    for (int s = 0; s < NSPLIT; ++s) {
        const size_t o = ((size_t)(b * NSPLIT + s)) * DX + d;
        sum += ws[o];
        sq  += ws[PART + o];
        mn   = fminf(mn, ws[2 * PART + o]);
        mx   = fmaxf(mx, ws[3 * PART + o]);
    }

    const float mean = sum * (1.0f / (float)NN);
    float var = (sq - sum * mean) * (1.0f / (float)(NN - 1));
    var = fmaxf(var, 0.0f);
    const float sd = sqrtf(var);

    float* __restrict__ zb = z + (size_t)b * DZ;
    zb[d]          = mean;
    zb[DX + d]     = mn;
    zb[2 * DX + d] = mx;
    zb[3 * DX + d] = sd;
}

// ---------------------------------------------------------------------------
// Kernel 3: out[64,256] = z[64,512] @ W^T + bias via V_WMMA_F32_16X16X4_F32.
// One wave per 16x16 output tile; 128 k-steps of K=4 (verified: emits 8
// v_wmma_f32_16x16x4_f32 per unrolled iteration with b64 operand loads).
//
// ISA VGPR layouts (CDNA5 ISA, 32-bit A 16x4 / C,D 16x16):
//   A: lanes 0-15 -> row M=lane, VGPR0=K0,VGPR1=K1 ; lanes 16-31 -> K2,K3
//   B: lanes 0-15 -> col N=lane, VGPR0=K0,VGPR1=K1 ; lanes 16-31 -> K2,K3
//   C/D: VGPR r -> M=r (lanes 0-15) / M=8+r (lanes 16-31), N=lane&15
// B[k][n] = W[ntile+n][k] since W is [DY, DZ] row-major (out = z @ W^T).
// ---------------------------------------------------------------------------
__global__ void __launch_bounds__(32)
wmma_gemm_kernel(const float* __restrict__ z,
                 const float* __restrict__ W,
                 const float* __restrict__ bias,
                 float* __restrict__ out)
{
    const int ntile = blockIdx.x << 4;   // 0..240
    const int mtile = blockIdx.y << 4;   // 0..48
    const int lane  = threadIdx.x;       // 0..31
    const int half  = lane >> 4;         // K pair selector
    const int l15   = lane & 15;         // M for A, N for B

    const float* __restrict__ arow = z + (size_t)(mtile + l15) * DZ + 2 * half;
    const float* __restrict__ brow = W + (size_t)(ntile + l15) * DZ + 2 * half;

    v8f c = {};

#pragma unroll 8
    for (int k = 0; k < DZ; k += 4) {
        v2f a = *(const v2f*)(arow + k);   // 8B-aligned: offset is even #floats
        v2f b = *(const v2f*)(brow + k);
        // (neg_a, A, neg_b, B, c_mod, C, reuse_a, reuse_b)
        c = __builtin_amdgcn_wmma_f32_16x16x4_f32(
                false, a, false, b, (short)0, c, false, false);
    }

    const float bv = bias[ntile + l15];

#pragma unroll
    for (int r = 0; r < 8; ++r) {
        const int m = (half << 3) + r;
        out[(size_t)(mtile + m) * DY + ntile + l15] = c[r] + bv;
    }
}

// ---------------------------------------------------------------------------
// Launch: X -> partials -> z (in ws) -> WMMA GEMM -> out
// ws usage: 4*PART floats of partials (8 MB) + BS*DZ floats of z (128 KB)
// ---------------------------------------------------------------------------
extern "C" void kernel_launch(void* const* d_in, const int* in_sizes, int n_in,
                              void* d_out, int out_size, void* d_ws, size_t ws_size,
                              hipStream_t stream)
{
    (void)in_sizes; (void)n_in; (void)out_size; (void)ws_size;

    const float* X  = (const float*)d_in[0];   // [64,16384,128]
    const float* W  = (const float*)d_in[1];   // [256,512]
    const float* bb = (const float*)d_in[2];   // [256]
    float*       out = (float*)d_out;          // [64,256]

    float* ws = (float*)d_ws;
    float* zb = ws + 4 * (size_t)PART;         // z: [64,512]

    partial_reduce_kernel<<<dim3(BS, NSPLIT), 32, 0, stream>>>(X, ws);
    combine_stats_kernel<<<dim3((BS * DX) / 128), 128, 0, stream>>>(ws, zb);
    wmma_gemm_kernel<<<dim3(DY / 16, BS / 16), 32, 0, stream>>>(zb, W, bb, out);
}